// VectorCommModule_48301202211078
// MI455X (gfx1250) — compile-verified
//
#include <hip/hip_runtime.h>
#include <hip/hip_bf16.h>
#include <math.h>

// ---------------- problem constants ----------------
#define HID 2048
#define BOT 64
#define CMM 128
#define NB  8
#define SEQ 4096
#define NSPLIT 64
#define ROWS_PER_SPLIT (SEQ / NSPLIT)   // 64

typedef float v2f __attribute__((ext_vector_type(2)));
typedef float v4f __attribute__((ext_vector_type(4)));
typedef float v8f __attribute__((ext_vector_type(8)));

// ---------------- WMMA f32 16x16x4 wrapper ----------------
// D = A(16x4) * B(4x16) + C(16x16), fp32 throughout.
__device__ __forceinline__ v8f wmma4(v2f a, v2f b, v8f c) {
  return __builtin_amdgcn_wmma_f32_16x16x4_f32(
      /*neg_a=*/false, a, /*neg_b=*/false, b,
      /*c_mod=*/(short)0, c, /*reuse_a=*/false, /*reuse_b=*/false);
}

// A fragment: row-major [16][lda], 16x4 tile starting at column k0.
// ISA layout: lane<16 holds M=lane, VGPR0/1 = K=0/1; lane>=16 holds K=2/3.
__device__ __forceinline__ v2f ldA(const float* __restrict__ A, int lda, int k0, int lane) {
  int m  = lane & 15;
  int kb = (lane >> 4) << 1;
  const float* p = A + m * lda + k0 + kb;
  v2f r; r.x = p[0]; r.y = p[1];
  return r;
}

// B fragment: row-major [K][ldb], 4x16 tile at (k0, n0).
// Layout (B rows striped across lanes): VGPR0 = rows K=0 (lanes 0-15) / K=2
// (lanes 16-31); VGPR1 = rows K=1 / K=3.
__device__ __forceinline__ v2f ldB(const float* __restrict__ B, int ldb, int k0, int n0, int lane) {
  int n  = lane & 15;
  int kb = (lane >> 4) << 1;
  const float* p = B + (k0 + kb) * ldb + n0 + n;
  v2f r; r.x = p[0]; r.y = p[ldb];
  return r;
}

// ---------------- kernel 1: split-K mean over sequence ----------------
// grid (NSPLIT, HID/1024, NB), block 256. Each block sums 64 rows x 1024 cols.
__global__ __launch_bounds__(256)
void pool_partial(const float* __restrict__ hid, float* __restrict__ partial) {
  int split = blockIdx.x;
  int hb    = blockIdx.y;
  int b     = blockIdx.z;
  int h4    = hb * 256 + threadIdx.x;                   // float4 col 0..511
  const v4f* base = (const v4f*)hid +
      (size_t)b * SEQ * (HID / 4) + (size_t)split * ROWS_PER_SPLIT * (HID / 4) + h4;
  v4f s = {0.f, 0.f, 0.f, 0.f};
  for (int r = 0; r < ROWS_PER_SPLIT; ++r) {
    v4f v = base[(size_t)r * (HID / 4)];
    s += v;
  }
  ((v4f*)partial)[((size_t)split * NB + b) * (HID / 4) + h4] = s;
}

// grid 128 blocks x 256: i in [0, 32768). Rows 8..15 of pooled zero-padded.
__global__ __launch_bounds__(256)
void pool_final(const float* __restrict__ partial, float* __restrict__ pooled) {
  int i = blockIdx.x * 256 + threadIdx.x;
  if (i < NB * HID) {
    float s = 0.f;
    for (int sp = 0; sp < NSPLIT; ++sp) s += partial[sp * (NB * HID) + i];
    pooled[i] = s * (1.0f / (float)SEQ);
  } else {
    pooled[i] = 0.f;   // pad rows 8..15 for WMMA M=16
  }
}

// ---------------- kernel 2: whole MLP chain, one workgroup ----------------
__global__ __launch_bounds__(256)
void mlp_kernel(const float* __restrict__ pooled,       // [16][2048], rows 8+ zero
                const float* __restrict__ ew1, const float* __restrict__ eb1,
                const float* __restrict__ eg,  const float* __restrict__ ebt,
                const float* __restrict__ ew2, const float* __restrict__ eb2,
                const float* __restrict__ dw1, const float* __restrict__ db1,
                const float* __restrict__ dg,  const float* __restrict__ dbt,
                const float* __restrict__ dw2, const float* __restrict__ db2,
                const float* __restrict__ edges,
                float* __restrict__ expanded)            // [8][2048]
{
  __shared__ float sE[17];
  __shared__ float sH1[16][BOT];
  __shared__ float sCM[16][CMM];
  __shared__ float sG2[16][BOT];
  __shared__ float sMu[8], sRs[8];
  __shared__ float sTmp[4][16][16];

  const int tid  = threadIdx.x;
  const int wave = tid >> 5;
  const int lane = tid & 31;

  if (tid < 17) sE[tid] = edges[tid];
  for (int i = tid; i < 8 * BOT; i += 256) sH1[8 + i / BOT][i % BOT] = 0.f;
  for (int i = tid; i < 8 * CMM; i += 256) sCM[8 + i / CMM][i % CMM] = 0.f;
  for (int i = tid; i < 8 * BOT; i += 256) sG2[8 + i / BOT][i % BOT] = 0.f;
  __syncthreads();

  // ---- GEMM1: h1 = pooled[16x2048] @ ew1[2048x64]; K halved across wave pairs
  {
    const int n0 = (wave & 3) * 16;
    const int kh = wave >> 2;
    v8f acc = {0.f, 0.f, 0.f, 0.f, 0.f, 0.f, 0.f, 0.f};
    for (int k0 = kh * 1024; k0 < kh * 1024 + 1024; k0 += 4)
      acc = wmma4(ldA(pooled, HID, k0, lane), ldB(ew1, BOT, k0, n0, lane), acc);
    const int n = lane & 15, rb = (lane >> 4) * 8;
    if (kh == 1)
      for (int r = 0; r < 8; ++r) sTmp[wave & 3][rb + r][n] = acc[r];
    __syncthreads();
    if (kh == 0)
      for (int r = 0; r < 8; ++r) {
        int row = rb + r;
        if (row < 8)
          sH1[row][n0 + n] = acc[r] + sTmp[wave & 3][row][n] + eb1[n0 + n];
      }
  }
  __syncthreads();

  // ---- LayerNorm + exact-erf GELU on sH1 rows 0..7
  if (tid < 8) {
    float s = 0.f;
    for (int j = 0; j < BOT; ++j) s += sH1[tid][j];
    float mu = s / (float)BOT;
    float v = 0.f;
    for (int j = 0; j < BOT; ++j) { float d = sH1[tid][j] - mu; v += d * d; }
    sMu[tid] = mu;
    sRs[tid] = rsqrtf(v / (float)BOT + 1e-5f);
  }
  __syncthreads();
  for (int i = tid; i < 8 * BOT; i += 256) {
    int r = i / BOT, c = i % BOT;
    float x = (sH1[r][c] - sMu[r]) * sRs[r] * eg[c] + ebt[c];
    sH1[r][c] = 0.5f * x * (1.f + erff(x * 0.70710678118654752f));
  }
  __syncthreads();

  // ---- GEMM2: comm = h1[16x64] @ ew2[64x128] (+bias, quantize)
  {
    const int n0 = wave * 16;
    v8f acc = {0.f, 0.f, 0.f, 0.f, 0.f, 0.f, 0.f, 0.f};
    for (int k0 = 0; k0 < BOT; k0 += 4)
      acc = wmma4(ldA(&sH1[0][0], BOT, k0, lane), ldB(ew2, CMM, k0, n0, lane), acc);
    const int n = lane & 15, rb = (lane >> 4) * 8;
    for (int r = 0; r < 8; ++r) {
      int row = rb + r;
      if (row < 8) {
        float x = acc[r] + eb2[n0 + n];
        float q;
        if (x <= sE[0])       q = sE[0];
        else if (x > sE[16])  q = sE[16];
        else {
          q = 0.f;
          for (int i2 = 1; i2 <= 16; ++i2)
            if (x <= sE[i2]) { q = 0.5f * (sE[i2 - 1] + sE[i2]); break; }
        }
        sCM[row][n0 + n] = q;
      }
    }
  }
  __syncthreads();

  // ---- GEMM3: h2 = comm[16x128] @ dw1[128x64]; K halved across wave pairs
  {
    const int n0 = (wave & 3) * 16;
    const int kh = wave >> 2;
    v8f acc = {0.f, 0.f, 0.f, 0.f, 0.f, 0.f, 0.f, 0.f};
    for (int k0 = kh * 64; k0 < kh * 64 + 64; k0 += 4)
      acc = wmma4(ldA(&sCM[0][0], CMM, k0, lane), ldB(dw1, BOT, k0, n0, lane), acc);
    const int n = lane & 15, rb = (lane >> 4) * 8;
    if (kh == 1)
      for (int r = 0; r < 8; ++r) sTmp[wave & 3][rb + r][n] = acc[r];
    __syncthreads();
    if (kh == 0)
      for (int r = 0; r < 8; ++r) {
        int row = rb + r;
        if (row < 8)
          sG2[row][n0 + n] = acc[r] + sTmp[wave & 3][row][n] + db1[n0 + n];
      }
  }
  __syncthreads();

  // ---- LayerNorm + GELU on sG2 rows 0..7
  if (tid < 8) {
    float s = 0.f;
    for (int j = 0; j < BOT; ++j) s += sG2[tid][j];
    float mu = s / (float)BOT;
    float v = 0.f;
    for (int j = 0; j < BOT; ++j) { float d = sG2[tid][j] - mu; v += d * d; }
    sMu[tid] = mu;
    sRs[tid] = rsqrtf(v / (float)BOT + 1e-5f);
  }
  __syncthreads();
  for (int i = tid; i < 8 * BOT; i += 256) {
    int r = i / BOT, c = i % BOT;
    float x = (sG2[r][c] - sMu[r]) * sRs[r] * dg[c] + dbt[c];
    sG2[r][c] = 0.5f * x * (1.f + erff(x * 0.70710678118654752f));
  }
  __syncthreads();

  // ---- GEMM4: expanded = g2[16x64] @ dw2[64x2048] (+bias); 16 N-tiles/wave
  for (int t = wave; t < HID / 16; t += 8) {
    const int n0 = t * 16;
    v8f acc = {0.f, 0.f, 0.f, 0.f, 0.f, 0.f, 0.f, 0.f};
    for (int k0 = 0; k0 < BOT; k0 += 4)
      acc = wmma4(ldA(&sG2[0][0], BOT, k0, lane), ldB(dw2, HID, k0, n0, lane), acc);
    const int n = n0 + (lane & 15), rb = (lane >> 4) * 8;
    for (int r = 0; r < 8; ++r) {
      int row = rb + r;
      if (row < 8) expanded[row * HID + n] = acc[r] + db2[n];
    }
  }
}

// ---------------- kernel 3: residual stream add ----------------
// Reads hidden_states in REVERSE global order: the tail (~192 MB) is still
// L2-resident from pool_partial's ascending stream, so up to ~75% of this
// second read is served from L2 instead of HBM. Output is written once and
// never re-read -> non-temporal store (TH=NT) so it doesn't evict those
// hot lines while we harvest them.
__global__ __launch_bounds__(256)
void add_kernel(const v4f* __restrict__ hid, const v4f* __restrict__ exp4,
                v4f* __restrict__ out) {
  const int total = NB * SEQ * (HID / 4);   // 16,777,216 float4
  for (int i = blockIdx.x * blockDim.x + threadIdx.x; i < total;
       i += gridDim.x * blockDim.x) {
    int j  = total - 1 - i;          // reverse stream
    int b  = j >> 21;                // perb = SEQ*(HID/4) = 2^21
    int h4 = j & (HID / 4 - 1);
    v4f e = exp4[(b << 9) + h4];     // 64 KB table, L2/L0 hot
    v4f x = hid[j];
    v4f o = x + 0.1f * e;
    __builtin_nontemporal_store(o, &out[j]);
  }
}

// ---------------- launcher ----------------
extern "C" void kernel_launch(void* const* d_in, const int* in_sizes, int n_in,
                              void* d_out, int out_size, void* d_ws, size_t ws_size,
                              hipStream_t stream) {
  const float* hid   = (const float*)d_in[0];
  const float* ew1   = (const float*)d_in[1];
  const float* eb1   = (const float*)d_in[2];
  const float* eg    = (const float*)d_in[3];
  const float* ebt   = (const float*)d_in[4];
  const float* ew2   = (const float*)d_in[5];
  const float* eb2   = (const float*)d_in[6];
  const float* dw1   = (const float*)d_in[7];
  const float* db1   = (const float*)d_in[8];
  const float* dg    = (const float*)d_in[9];
  const float* dbt   = (const float*)d_in[10];
  const float* dw2   = (const float*)d_in[11];
  const float* db2   = (const float*)d_in[12];
  const float* edges = (const float*)d_in[13];

  float* pooled   = (float*)d_ws;             // 16*2048 floats (rows 8+ zero)
  float* expanded = pooled + 16 * HID;        // 8*2048 floats
  float* partial  = expanded + 8 * HID;       // NSPLIT * 8 * 2048 floats

  dim3 gp(NSPLIT, HID / 1024, NB);
  pool_partial<<<gp, 256, 0, stream>>>(hid, partial);
  pool_final<<<(16 * HID) / 256, 256, 0, stream>>>(partial, pooled);
  mlp_kernel<<<1, 256, 0, stream>>>(pooled, ew1, eb1, eg, ebt, ew2, eb2,
                                    dw1, db1, dg, dbt, dw2, db2, edges, expanded);
  add_kernel<<<16384, 256, 0, stream>>>((const v4f*)hid, (const v4f*)expanded,
                                        (v4f*)d_out);
}